// TrigunaNeuron_23330262352056
// MI455X (gfx1250) — compile-verified
//
#include <hip/hip_runtime.h>

// TrigunaNeuron: adaptive-threshold LIF scan.
// x: [T=256, B=32, F=2048] f32.  out: spikes [T,B,F] ++ man_history [T,B,F].
// Memory-bound (192 MB @ 23.3 TB/s ~ 8.3 us). Strategy: deep async
// global->LDS pipeline (ASYNCcnt) + non-temporal streaming stores.

typedef float v4f __attribute__((ext_vector_type(4)));
typedef int   v4i __attribute__((vector_size(16)));   // matches builtin param type

#define T_STEPS 256
#define DECAY   0.9f
#define BASE_TH 1.0f
#define TAPASYA 0.05f
#define RELAX   0.005f

// gfx1250 async tensor-path builtin availability (device pass only).
#if defined(__HIP_DEVICE_COMPILE__) && __has_builtin(__builtin_amdgcn_global_load_async_to_lds_b128)
#define USE_ASYNC 1
#else
#define USE_ASYNC 0
#endif

#if USE_ASYNC
typedef __attribute__((address_space(1))) v4i* g_v4i_ptr;   // global (AS1)
typedef __attribute__((address_space(3))) v4i* l_v4i_ptr;   // LDS (AS3)
#endif

__device__ __forceinline__ void lif_step(const v4f& cur, v4f& man, v4f& dyn, v4f& spike) {
#pragma unroll
  for (int i = 0; i < 4; ++i) {
    float m = DECAY * man[i] + cur[i];           // leaky integrate
    float s = (m >= dyn[i]) ? 1.0f : 0.0f;       // spike vs pre-update threshold
    m = m - s * dyn[i];                          // soft reset
    float d = dyn[i] + s * TAPASYA;              // raise on spike
    d = d - (d - BASE_TH) * RELAX;               // homeostatic relaxation
    man[i] = m; dyn[i] = d; spike[i] = s;
  }
}

__global__ __launch_bounds__(128) void
triguna_scan_kernel(const float* __restrict__ x, float* __restrict__ out, int bf) {
  const int tid = threadIdx.x;
  const int g   = blockIdx.x * blockDim.x + tid;     // float4 stream id
  const size_t col = (size_t)g * 4;
  if ((int)col >= bf) return;                        // exact fit: 16384 * 4 == bf

  const float* xg     = x + col;                     // per-step stride = bf floats
  float* __restrict__ spikes = out;                  // [T, bf]
  float* __restrict__ manh   = out + (size_t)T_STEPS * bf;

  v4f man   = {0.0f, 0.0f, 0.0f, 0.0f};
  v4f dyn   = {BASE_TH, BASE_TH, BASE_TH, BASE_TH};
  v4f spike;

#if USE_ASYNC
  // ---- CDNA5 path: 32-deep async global->LDS ring, ASYNCcnt-ordered ----
  constexpr int STAGES = 32;                         // 32 * 128 thr * 16B = 64KB LDS
  __shared__ v4f stage[STAGES * 128];

  // Prologue: fill the ring.
#pragma unroll 4
  for (int s = 0; s < STAGES; ++s) {
    __builtin_amdgcn_global_load_async_to_lds_b128(
        (g_v4i_ptr)(xg + (size_t)s * bf),
        (l_v4i_ptr)&stage[s * 128 + tid], 0, 0);
  }

  for (int t = 0; t < T_STEPS; ++t) {
    // Async loads complete in order; <=31 outstanding => op #t has landed.
    asm volatile("s_wait_asynccnt 31" ::: "memory");
    const int slot = (t & (STAGES - 1)) * 128 + tid;
    v4f cur = stage[slot];
    // Ensure the ds_read drained before the async refill can overwrite the slot.
    asm volatile("s_wait_dscnt 0" ::: "memory");
    // Always refill (clamped address in the tail) to keep the count invariant.
    int tn = t + STAGES; tn = (tn < T_STEPS) ? tn : (T_STEPS - 1);
    __builtin_amdgcn_global_load_async_to_lds_b128(
        (g_v4i_ptr)(xg + (size_t)tn * bf),
        (l_v4i_ptr)&stage[slot], 0, 0);

    lif_step(cur, man, dyn, spike);

    __builtin_nontemporal_store(spike, (v4f*)(spikes + (size_t)t * bf + col));
    __builtin_nontemporal_store(man,   (v4f*)(manh   + (size_t)t * bf + col));
  }
#else
  // ---- Fallback: register double-buffer + global_prefetch_b8 lookahead ----
  constexpr int PF = 24;                             // cachelines ~24 steps ahead
  v4f nxt = __builtin_nontemporal_load((const v4f*)xg);
  for (int t = 0; t < T_STEPS; ++t) {
    v4f cur = nxt;
    int tn = (t + 1 < T_STEPS) ? t + 1 : (T_STEPS - 1);
    nxt = __builtin_nontemporal_load((const v4f*)(xg + (size_t)tn * bf));
    if (t + PF < T_STEPS)
      __builtin_prefetch(xg + (size_t)(t + PF) * bf, 0, 0);

    lif_step(cur, man, dyn, spike);

    __builtin_nontemporal_store(spike, (v4f*)(spikes + (size_t)t * bf + col));
    __builtin_nontemporal_store(man,   (v4f*)(manh   + (size_t)t * bf + col));
  }
#endif
}

extern "C" void kernel_launch(void* const* d_in, const int* in_sizes, int n_in,
                              void* d_out, int out_size, void* d_ws, size_t ws_size,
                              hipStream_t stream) {
  (void)n_in; (void)out_size; (void)d_ws; (void)ws_size;
  const float* x = (const float*)d_in[0];
  float* out = (float*)d_out;

  const int bf = in_sizes[0] / T_STEPS;              // B*F = 65536
  const int nstreams = bf / 4;                        // 16384 float4 streams
  dim3 block(128);
  dim3 grid((nstreams + block.x - 1) / block.x);      // 128 blocks

  triguna_scan_kernel<<<grid, block, 0, stream>>>(x, out, bf);
}